// LearnableTD_12343736009203
// MI455X (gfx1250) — compile-verified
//
#include <hip/hip_runtime.h>
#include <hip/hip_bf16.h>

// TD(lambda) backward affine scan, segmented for MI455X (gfx1250).
//   y_t = b_t + a_t * y_{t+1},  a_t = gamma*(1-done_t)*lambda_t
// Input tiles are staged global->LDS with the CDNA5 Tensor Data Mover (TDM):
// one tensor_load_to_lds per 32x32 strided tile, LDS row padding done by the
// DMA engine (pad 1 DWORD per 32 -> stride 33 floats, bank-conflict free),
// double-buffered on TENSORcnt.

#define S_SEG 32   // time segments
#define CT    32   // time steps per staged chunk
#define RPW   32   // batch rows per wave (= wave32 lanes)
#define PAD   33   // LDS row stride in floats (TDM pad: +1 DWORD per 32 DWORDs)

typedef __attribute__((ext_vector_type(4))) unsigned u32x4;
typedef __attribute__((ext_vector_type(8))) int      i32x8;
typedef __attribute__((ext_vector_type(4))) int      i32x4;

__device__ __forceinline__ float sig_m1p1(float x) {
  // 2*sigmoid(x) - 1 == tanh(x/2)
  float e = __expf(-x);
  return (1.0f - e) / (1.0f + e);
}

__device__ __forceinline__ unsigned lds_addr32(const void* p) {
  // flat LDS pointer: low 32 bits are the LDS byte offset (ISA 10.2 aperture rules)
  return (unsigned)(unsigned long long)p;
}

// ---- TDM: DMA one 32x32 f32 tile (row stride = stride_elems) into LDS -------
// D# layout per CDNA5 ISA 8.3/8.4. LDS padding: pad_enable=1, pad_interval
// code 4 (pad after every 32 DWORDs), pad_amount code 0 (pad = 1 DWORD)
// -> LDS row stride = 33 floats.
__device__ __forceinline__ void tdm_load_tile(const float* gaddr, unsigned lds_byte,
                                              unsigned stride_elems) {
  const unsigned long long ga = (unsigned long long)gaddr;
  u32x4 g0;
  g0[0] = 1u;                                          // count=1 (valid, user mode)
  g0[1] = lds_byte;                                    // lds_addr (bytes)
  g0[2] = (unsigned)(ga & 0xFFFFFFFFu);                // global_addr[31:0]
  g0[3] = (unsigned)((ga >> 32) & 0x01FFFFFFu)         // global_addr[56:32]
        | (2u << 30);                                  // type = 2 ("image")
  i32x8 g1;
  g1[0] = (int)((2u << 16) | (1u << 20) | (4u << 22)); // data_size=4B, pad_en, pad/32DW, pad=1DW
  g1[1] = (int)(32u << 16);                            // tensor_dim0 = 32 (lo16)
  g1[2] = (int)(32u << 16);                            // dim0 hi=0 | tensor_dim1 = 32 (lo16)
  g1[3] = (int)(32u << 16);                            // dim1 hi=0 | tile_dim0 = 32
  g1[4] = 32;                                          // tile_dim1 = 32, tile_dim2 = 0
  g1[5] = (int)stride_elems;                           // tensor_dim0_stride[31:0]
  g1[6] = 0;                                           // stride hi | tensor_dim1_stride lo
  g1[7] = 0;
#if __has_builtin(__builtin_amdgcn_tensor_load_to_lds)
  // amdgpu-toolchain (clang-23) 6-arg form:
  //   (uint32x4 g0, int32x8 g1, int32x4, int32x4, int32x8, i32 cpol)
  // 2D tensor (tile_dim2 = 0): trailing descriptor groups are don't-cares -> zeros.
  i32x4 gz4 = {0, 0, 0, 0};
  i32x8 gz8 = {0, 0, 0, 0, 0, 0, 0, 0};
  __builtin_amdgcn_tensor_load_to_lds(g0, g1, gz4, gz4, gz8, 0);
#else
  asm volatile("tensor_load_to_lds %0, %1" :: "s"(g0), "s"(g1) : "memory");
#endif
}

__device__ __forceinline__ void wait_tensor0() {
#if __has_builtin(__builtin_amdgcn_s_wait_tensorcnt)
  __builtin_amdgcn_s_wait_tensorcnt((short)0);
#else
  asm volatile("s_wait_tensorcnt 0x0" ::: "memory");
#endif
}
__device__ __forceinline__ void wait_tensor_prev() {
  // TDM ops complete in-order per wave: cnt<=3 => the older chunk's 3 tiles landed
#if __has_builtin(__builtin_amdgcn_s_wait_tensorcnt)
  __builtin_amdgcn_s_wait_tensorcnt((short)3);
#else
  asm volatile("s_wait_tensorcnt 0x3" ::: "memory");
#endif
}
__device__ __forceinline__ void wait_ds0() {
  asm volatile("s_wait_dscnt 0x0" ::: "memory");
}

// Issue (no wait) the three 32x32 tiles of one chunk via TDM.
__device__ __forceinline__ void issue_chunk_tdm(
    const float* __restrict__ rewards, const float* __restrict__ dones,
    const float* __restrict__ values,
    float* tr, float* td, float* tv, int r0, int t0, int T)
{
  tdm_load_tile(rewards + (size_t)r0 * T + t0,             lds_addr32(tr), (unsigned)T);
  tdm_load_tile(dones   + (size_t)r0 * T + t0,             lds_addr32(td), (unsigned)T);
  tdm_load_tile(values  + (size_t)r0 * (T + 1) + t0 + 1,   lds_addr32(tv), (unsigned)(T + 1));
}

// Precompute dynamic lambda per timestep into workspace.
__global__ void td_prep(const float* __restrict__ raw_lambd,
                        const int* __restrict__ start_idx,
                        float* __restrict__ lam_ws, int T) {
  int t = blockIdx.x * blockDim.x + threadIdx.x;
  if (t < T) lam_ws[t] = 0.95f + 0.05f * sig_m1p1(raw_lambd[start_idx[0] + t]);
}

// Pass 1: per (row, segment): alpha = prod(a), zero-carry partials for lam & sum.
__global__ __launch_bounds__(RPW) void td_pass1(
    const float* __restrict__ rewards, const float* __restrict__ dones,
    const float* __restrict__ values,  const float* __restrict__ lam_ws,
    const float* __restrict__ raw_gamma,
    float* __restrict__ alpha, float* __restrict__ blam, float* __restrict__ bsum,
    int B, int T, int L)
{
  __shared__ alignas(16) float tr[2][RPW * PAD], td_[2][RPW * PAD], tv[2][RPW * PAD];
  const int lane = threadIdx.x;
  const int s    = blockIdx.x % S_SEG;
  const int r0   = (blockIdx.x / S_SEG) * RPW;
  const int b    = r0 + lane;
  const float g  = 0.99f + 0.01f * sig_m1p1(raw_gamma[0]);
  const int NC   = L / CT;

  float P = 1.0f, bl = 0.0f, bs = 0.0f;
  int k = (NC - 1) & 1;
  issue_chunk_tdm(rewards, dones, values, tr[k], td_[k], tv[k], r0, s * L + (NC - 1) * CT, T);
  float lv_cur = lam_ws[s * L + (NC - 1) * CT + lane];   // lambda chunk, one per lane
  float lv_nxt = 0.0f;

  for (int c = NC - 1; c >= 0; --c) {
    k = c & 1;
    if (c > 0) {  // prefetch next (older-in-time) chunk into the other buffer
      wait_ds0();
      issue_chunk_tdm(rewards, dones, values, tr[k ^ 1], td_[k ^ 1], tv[k ^ 1],
                      r0, s * L + (c - 1) * CT, T);
      lv_nxt = lam_ws[s * L + (c - 1) * CT + lane];
      wait_tensor_prev();
    } else {
      wait_tensor0();
    }
#pragma unroll
    for (int tt = CT - 1; tt >= 0; --tt) {
      float r  = tr[k][lane * PAD + tt];
      float d  = td_[k][lane * PAD + tt];
      float vn = tv[k][lane * PAD + tt];
      float l  = __shfl(lv_cur, tt);   // uniform index -> v_readlane broadcast
      float cont = g * (1.0f - d);
      float a = cont * l;
      bl = r + cont * (1.0f - l) * vn + a * bl;
      bs = r + a * bs;
      P *= a;
    }
    lv_cur = lv_nxt;
  }
  const size_t idx = (size_t)s * B + b;
  alpha[idx] = P; blam[idx] = bl; bsum[idx] = bs;
}

// Pass 2a: per-row carry propagation across segments; also writes lambda_returns[:, T].
__global__ void td_pass2a(const float* __restrict__ values,
                          const float* __restrict__ alpha, const float* __restrict__ blam,
                          const float* __restrict__ bsum,
                          float* __restrict__ clam, float* __restrict__ csum,
                          float* __restrict__ out_lam, int B, int T)
{
  int b = blockIdx.x * blockDim.x + threadIdx.x;
  if (b >= B) return;
  float vlast = values[(size_t)b * (T + 1) + T];
  out_lam[(size_t)b * (T + 1) + T] = vlast;
  float cl = vlast, cs = 0.0f;
  for (int s = S_SEG - 1; s >= 0; --s) {
    size_t idx = (size_t)s * B + b;
    clam[idx] = cl; csum[idx] = cs;
    float A = alpha[idx];
    cl = blam[idx] + A * cl;
    cs = bsum[idx] + A * cs;
  }
}

// Pass 2b: re-scan each segment with its true carry; outputs staged in LDS, flushed coalesced.
__global__ __launch_bounds__(RPW) void td_pass2b(
    const float* __restrict__ rewards, const float* __restrict__ dones,
    const float* __restrict__ values,  const float* __restrict__ lam_ws,
    const float* __restrict__ raw_gamma,
    const float* __restrict__ clam, const float* __restrict__ csum,
    float* __restrict__ out_lam, float* __restrict__ out_sum,
    int B, int T, int L)
{
  __shared__ alignas(16) float tr[2][RPW * PAD], td_[2][RPW * PAD], tv[2][RPW * PAD];
  __shared__ alignas(16) float ol[RPW * PAD], os_[RPW * PAD];
  const int lane = threadIdx.x;
  const int s    = blockIdx.x % S_SEG;
  const int r0   = (blockIdx.x / S_SEG) * RPW;
  const int b    = r0 + lane;
  const float g  = 0.99f + 0.01f * sig_m1p1(raw_gamma[0]);
  const int NC   = L / CT;

  float lam = clam[(size_t)s * B + b];
  float sum = csum[(size_t)s * B + b];
  int k = (NC - 1) & 1;
  issue_chunk_tdm(rewards, dones, values, tr[k], td_[k], tv[k], r0, s * L + (NC - 1) * CT, T);
  float lv_cur = lam_ws[s * L + (NC - 1) * CT + lane];
  float lv_nxt = 0.0f;

  for (int c = NC - 1; c >= 0; --c) {
    k = c & 1;
    const int t0 = s * L + c * CT;
    if (c > 0) {
      wait_ds0();
      issue_chunk_tdm(rewards, dones, values, tr[k ^ 1], td_[k ^ 1], tv[k ^ 1],
                      r0, s * L + (c - 1) * CT, T);
      lv_nxt = lam_ws[s * L + (c - 1) * CT + lane];
      wait_tensor_prev();
    } else {
      wait_tensor0();
    }
#pragma unroll
    for (int tt = CT - 1; tt >= 0; --tt) {
      float r  = tr[k][lane * PAD + tt];
      float d  = td_[k][lane * PAD + tt];
      float vn = tv[k][lane * PAD + tt];
      float l  = __shfl(lv_cur, tt);
      float cont = g * (1.0f - d);
      float a = cont * l;
      sum = r + a * sum;
      lam = r + cont * (1.0f - l) * vn + a * lam;
      ol[lane * PAD + tt] = lam;
      os_[lane * PAD + tt] = sum;
    }
    lv_cur = lv_nxt;
    // coalesced flush: lane = time offset, iterate rows (same-wave ds ordering is in-order)
#pragma unroll
    for (int i = 0; i < RPW; ++i) {
      out_lam[(size_t)(r0 + i) * (T + 1) + t0 + lane] = ol[i * PAD + lane];
      out_sum[(size_t)(r0 + i) * T + t0 + lane]       = os_[i * PAD + lane];
    }
  }
}

// Fallback: one thread per row, full sequential scan (used only if ws too small / odd shapes).
__global__ void td_naive(const float* __restrict__ rewards, const float* __restrict__ dones,
                         const float* __restrict__ values, const float* __restrict__ raw_lambd,
                         const float* __restrict__ raw_gamma, const int* __restrict__ start_idx,
                         float* __restrict__ out_lam, float* __restrict__ out_sum, int B, int T)
{
  int b = blockIdx.x * blockDim.x + threadIdx.x;
  if (b >= B) return;
  const int s0 = start_idx[0];
  const float g = 0.99f + 0.01f * sig_m1p1(raw_gamma[0]);
  float vlast = values[(size_t)b * (T + 1) + T];
  out_lam[(size_t)b * (T + 1) + T] = vlast;
  float lam = vlast, sum = 0.0f;
  for (int t = T - 1; t >= 0; --t) {
    float l = 0.95f + 0.05f * sig_m1p1(raw_lambd[s0 + t]);
    float r = rewards[(size_t)b * T + t];
    float d = dones[(size_t)b * T + t];
    float vn = values[(size_t)b * (T + 1) + t + 1];
    float cont = g * (1.0f - d);
    float a = cont * l;
    sum = r + a * sum;
    lam = r + cont * (1.0f - l) * vn + a * lam;
    out_sum[(size_t)b * T + t] = sum;
    out_lam[(size_t)b * (T + 1) + t] = lam;
  }
}

extern "C" void kernel_launch(void* const* d_in, const int* in_sizes, int n_in,
                              void* d_out, int out_size, void* d_ws, size_t ws_size,
                              hipStream_t stream) {
  const float* values    = (const float*)d_in[0];
  const float* rewards   = (const float*)d_in[1];
  const float* dones     = (const float*)d_in[2];
  const float* raw_gamma = (const float*)d_in[3];
  const float* raw_lambd = (const float*)d_in[4];
  const int*   start_idx = (const int*)d_in[5];
  (void)n_in; (void)out_size;

  const int T = in_sizes[4];
  const int B = in_sizes[0] / (T + 1);
  float* out_lam = (float*)d_out;
  float* out_sum = out_lam + (size_t)B * (T + 1);

  const int L = T / S_SEG;
  const bool shapes_ok = (T % (S_SEG * CT) == 0) && (B % RPW == 0);
  const size_t need = sizeof(float) * ((size_t)T + 5ull * (size_t)S_SEG * (size_t)B);

  if (shapes_ok && ws_size >= need) {
    float* w      = (float*)d_ws;
    float* lam_ws = w;
    float* alpha  = lam_ws + T;
    float* blam   = alpha + (size_t)S_SEG * B;
    float* bsum   = blam  + (size_t)S_SEG * B;
    float* clam   = bsum  + (size_t)S_SEG * B;
    float* csum   = clam  + (size_t)S_SEG * B;

    td_prep<<<(T + 255) / 256, 256, 0, stream>>>(raw_lambd, start_idx, lam_ws, T);
    td_pass1<<<(B / RPW) * S_SEG, RPW, 0, stream>>>(
        rewards, dones, values, lam_ws, raw_gamma, alpha, blam, bsum, B, T, L);
    td_pass2a<<<(B + 255) / 256, 256, 0, stream>>>(
        values, alpha, blam, bsum, clam, csum, out_lam, B, T);
    td_pass2b<<<(B / RPW) * S_SEG, RPW, 0, stream>>>(
        rewards, dones, values, lam_ws, raw_gamma, clam, csum, out_lam, out_sum, B, T, L);
  } else {
    td_naive<<<(B + 255) / 256, 256, 0, stream>>>(
        rewards, dones, values, raw_lambd, raw_gamma, start_idx, out_lam, out_sum, B, T);
  }
}